// MaskingModule_18064632447413
// MI455X (gfx1250) — compile-verified
//
#include <hip/hip_runtime.h>
#include <cstdint>

// -----------------------------------------------------------------------------
// MaskingModule on MI455X (gfx1250):
//   out[b,s,:] = mask[b,s] ? mask_item[:] : seq[b,s,:]
//   mask = dilated Bernoulli(0.065) (threefry2x32, key {0,42}), window [s-5,s+4],
//          last position uses the raw Bernoulli sample.
// Pure HBM-bound streaming select: ~134MB in (49% skipped) + 134MB out.
// CDNA5 paths: global_load_async_to_lds_b128 (ASYNCcnt) staging of mask_item,
// s_wait_asynccnt, NT temporal hints on b128 streaming traffic, wave32 blocks.
// -----------------------------------------------------------------------------

typedef float v4f __attribute__((ext_vector_type(4)));

#define B_DIM 16
#define S_DIM 4096
#define C_DIM 512
#define N_ROWS (B_DIM * S_DIM)       // 65536
#define HALF_N (N_ROWS / 2)          // 32768

__device__ __forceinline__ uint32_t rotl32(uint32_t x, uint32_t n) {
  return (x << n) | (x >> (32u - n));
}

// jax threefry2x32 with key (k0=0, k1=42): 20 rounds, 5 key-injection steps.
__device__ __forceinline__ void threefry2x32_0_42(uint32_t c0, uint32_t c1,
                                                  uint32_t& o0, uint32_t& o1) {
  const uint32_t k0 = 0u;
  const uint32_t k1 = 42u;
  const uint32_t k2 = k0 ^ k1 ^ 0x1BD11BDAu;
  uint32_t x0 = c0 + k0;
  uint32_t x1 = c1 + k1;
#define TF_R(r) { x0 += x1; x1 = rotl32(x1, (r)); x1 ^= x0; }
  TF_R(13) TF_R(15) TF_R(26) TF_R(6)
  x0 += k1; x1 += k2 + 1u;
  TF_R(17) TF_R(29) TF_R(16) TF_R(24)
  x0 += k2; x1 += k0 + 2u;
  TF_R(13) TF_R(15) TF_R(26) TF_R(6)
  x0 += k0; x1 += k1 + 3u;
  TF_R(17) TF_R(29) TF_R(16) TF_R(24)
  x0 += k1; x1 += k2 + 4u;
  TF_R(13) TF_R(15) TF_R(26) TF_R(6)
  x0 += k2; x1 += k0 + 5u;
#undef TF_R
  o0 = x0; o1 = x1;
}

// jax.random.uniform < p : u = bitcast(bits>>9 | 0x3F800000) - 1.0f
__device__ __forceinline__ unsigned char bern065(uint32_t bits) {
  float u = __uint_as_float((bits >> 9) | 0x3F800000u) - 1.0f;
  return (u < 0.065f) ? (unsigned char)1 : (unsigned char)0;
}

// Kernel A: materialize the 64KB sparse Bernoulli bitmap.
// jax splits the flat 65536-counter iota into halves: pair (i, i+32768).
__global__ void sparse_bits_kernel(unsigned char* __restrict__ sparse) {
  const uint32_t i = blockIdx.x * blockDim.x + threadIdx.x;  // 0..32767
  uint32_t a, b;
  threefry2x32_0_42(i, i + HALF_N, a, b);
  sparse[i] = bern065(a);
  sparse[i + HALF_N] = bern065(b);
}

// Kernel B: dilation + masked scatter. 256 threads (8 wave32s) per block,
// 2 rows per block, one float4 per lane per row.
__global__ void __launch_bounds__(256)
mask_scatter_kernel(const float* __restrict__ seq,
                    const float* __restrict__ mask_item,
                    const unsigned char* __restrict__ sparse,
                    float* __restrict__ out,
                    int* __restrict__ mask_out) {
  __shared__ __align__(16) float smem[C_DIM];
  __shared__ int mrow[2];

  const int t = threadIdx.x;
  const int row0 = (int)(blockIdx.x << 1);

  // --- CDNA5 async tensor path: stage mask_item (2KB) global -> LDS ---------
  if (t < C_DIM / 4) {
    uint64_t gaddr = (uint64_t)(uintptr_t)(mask_item + t * 4);
    // flat LDS pointers carry the wave-relative LDS offset in the low 32 bits
    uint32_t laddr = (uint32_t)(uintptr_t)(&smem[t * 4]);
    asm volatile("global_load_async_to_lds_b128 %0, %1, off"
                 :: "v"(laddr), "v"(gaddr)
                 : "memory");
  }

  // --- per-row dilated mask (window [s-5, s+4]; last position = raw sample) -
  if (t < 2) {
    const int rid = row0 + t;
    const int s = rid & (S_DIM - 1);
    const unsigned char* sp = sparse + (rid - s);  // row base: b * S_DIM
    int m;
    if (s == S_DIM - 1) {
      m = (int)sp[s];
    } else {
      const int lo = (s - 5 < 0) ? 0 : (s - 5);
      const int hi = (s + 4 > S_DIM - 1) ? (S_DIM - 1) : (s + 4);
      m = 0;
      for (int j = lo; j <= hi; ++j) m |= (int)sp[j];
    }
    mrow[t] = m;
    mask_out[rid] = m;  // second reference output (int32 dense mask)
  }

  asm volatile("s_wait_asynccnt 0" ::: "memory");
  __syncthreads();

  // --- streaming select: one b128 per lane, NT hints (268MB > 192MB L2) -----
  const int r = t >> 7;         // which of the 2 rows
  const int lane = t & 127;     // float4 index within the 512-float channel
  const long rid = (long)(row0 + r);

  const v4f* src = (const v4f*)seq + rid * (C_DIM / 4) + lane;
  v4f*       dst = (v4f*)out       + rid * (C_DIM / 4) + lane;

  v4f v;
  if (mrow[r]) {
    // masked row: broadcast embedding from LDS, skip the global read entirely
    v = ((const v4f*)smem)[lane];
  } else {
    v = __builtin_nontemporal_load(src);
  }
  __builtin_nontemporal_store(v, dst);
}

extern "C" void kernel_launch(void* const* d_in, const int* in_sizes, int n_in,
                              void* d_out, int out_size, void* d_ws, size_t ws_size,
                              hipStream_t stream) {
  const float* seq       = (const float*)d_in[0];  // [16, 4096, 512] f32
  const float* mask_item = (const float*)d_in[1];  // [1, 512] f32

  unsigned char* sparse = (unsigned char*)d_ws;    // 64 KB scratch bitmap

  float* out     = (float*)d_out;                                   // output 0
  int*   mask_out = (int*)((float*)d_out + (size_t)N_ROWS * C_DIM); // output 1

  sparse_bits_kernel<<<HALF_N / 256, 256, 0, stream>>>(sparse);
  mask_scatter_kernel<<<N_ROWS / 2, 256, 0, stream>>>(seq, mask_item, sparse,
                                                      out, mask_out);
}